// FourierInterpolator_49916109914678
// MI455X (gfx1250) — compile-verified
//
#include <hip/hip_runtime.h>
#include <hip/hip_bf16.h>
#include <math.h>

// ---------------------------------------------------------------------------
// Fourier interpolation, factorized into a WMMA GEMM.
//
//   out[b,n] = (1/1024) * sum_{i,j} Fr[b,i,j]*cos(phi) - Fi[b,i,j]*sin(phi)
//   phi      = 2*pi*(i*u_n + j*v_n),  u = xnew[...,0]*31/32, v = xnew[...,1]*31/32
//
// Separating the phase:  out = sum_i Cu[i]*A[n,i] - Su[i]*B[n,i]  with
//   [A|B] = E * M,   E = [cos(2pi j v) | sin(2pi j v)]  (N x 64)
//   M = [[Fr^T, Fi^T], [-Fi^T, Fr^T]]                   (64 x 64, per batch)
//
// GEMM (per batch 4096 x 64 x 64, fp32) -> V_WMMA_F32_16X16X4_F32.
// M is staged *transposed* in LDS so each B fragment (two K-consecutive
// values per lane) is one aligned ds_load_b64 into an even VGPR pair.
// ---------------------------------------------------------------------------

typedef float v2f __attribute__((ext_vector_type(2)));
typedef float v8f __attribute__((ext_vector_type(8)));

#define TWO_PI 6.28318530717958647692f

// ---------------------------------------------------------------------------
// Kernel 1: 32x32 DFT of y per batch (separable, LDS two-stage), then emit the
// 64x64 matrix M[b] = [[Fr^T, Fi^T], [-Fi^T, Fr^T]] into workspace.
// One block per batch, thread (i,j) owns one frequency bin.
// ---------------------------------------------------------------------------
__global__ __launch_bounds__(1024) void fi_fft_build_M(const float* __restrict__ y,
                                                       float* __restrict__ M) {
  const int b   = blockIdx.x;
  const int tid = threadIdx.x;            // 0..1023

  __shared__ float ysh[1024];
  __shared__ float Gr[1024];
  __shared__ float Gi[1024];
  __shared__ float ct[32];
  __shared__ float st[32];

  ysh[tid] = y[b * 1024 + tid];
  if (tid < 32) {
    float a = TWO_PI * (float)tid * (1.0f / 32.0f);
    ct[tid] = __cosf(a);
    st[tid] = __sinf(a);
  }
  __syncthreads();

  // Stage 1: DFT along dim0 (p).  tid -> (i, q):  G[i,q] = sum_p y[p,q] e^{-2pi i*p/32}
  {
    const int i = tid >> 5, q = tid & 31;
    float ar = 0.0f, ai = 0.0f;
    #pragma unroll 4
    for (int p = 0; p < 32; ++p) {
      const float v = ysh[p * 32 + q];
      const int   k = (i * p) & 31;
      ar += v * ct[k];
      ai -= v * st[k];
    }
    Gr[i * 32 + q] = ar;
    Gi[i * 32 + q] = ai;
  }
  __syncthreads();

  // Stage 2: DFT along dim1 (q).  tid -> (i, j):  F[i,j] = sum_q G[i,q] e^{-2pi j*q/32}
  const int i = tid >> 5, j = tid & 31;
  float fr = 0.0f, fi = 0.0f;
  #pragma unroll 4
  for (int q = 0; q < 32; ++q) {
    const int   k = (j * q) & 31;
    const float c = ct[k], s = st[k];
    const float gr = Gr[i * 32 + q], gi = Gi[i * 32 + q];
    fr += gr * c + gi * s;          // (Gr + iGi)(c - is) real
    fi += gi * c - gr * s;          // (Gr + iGi)(c - is) imag
  }

  float* Mb = M + b * 4096;
  Mb[j * 64 + i]              =  fr;   // A-columns, Cv rows
  Mb[j * 64 + 32 + i]         =  fi;   // B-columns, Cv rows
  Mb[(32 + j) * 64 + i]       = -fi;   // A-columns, Sv rows
  Mb[(32 + j) * 64 + 32 + i]  =  fr;   // B-columns, Sv rows
}

// ---------------------------------------------------------------------------
// Kernel 2: per (batch, 128-row strip):
//   - stage M^T (64x64, stride 68) in LDS: B-fragment K-pairs are contiguous
//   - each of 8 waves builds its 16x64 E tile (stride 68) in a private region
//   - P(16x64) = E * M via 16 K-steps x 4 N-tiles of v_wmma_f32_16x16x4_f32,
//     fully unrolled; A and B fragments are single aligned ds_load_b64 ops
//   - epilogue: out[n] = (1/1024) * sum_i Cu[i]*P[n,i] - Su[i]*P[n,32+i],
//     split across lane pairs (lane, lane^16), combined with shfl_xor.
//
// Stride 68: even (8B-aligned v2f loads) and bank-clean: lane n of a b64 load
// touches banks {4n+k, 4n+k+1}; the +2 K-offset of lanes 16..31 lands on
// {4n+k+2, 4n+k+3} -> all 32 lanes on distinct banks.
// ---------------------------------------------------------------------------
#define ES 68   // padded row stride (floats) for E/P tiles and M^T

__global__ __launch_bounds__(256) void fi_interp_wmma(const float* __restrict__ xnew,
                                                      const float* __restrict__ M,
                                                      float* __restrict__ out) {
  const int b       = blockIdx.y;
  const int rowBase = blockIdx.x * 128;          // 128 sample rows per block
  const int wave    = threadIdx.x >> 5;          // 0..7
  const int lane    = threadIdx.x & 31;
  const int waveRow = rowBase + wave * 16;       // first of this wave's 16 rows

  __shared__ float MshT[64 * ES];                // ~17 KB, M transposed: MshT[n][k] = M[k][n]
  __shared__ float Esh[8][16 * ES];              // ~34 KB, wave-private tiles

  // Stage M transposed for this batch. idx = k*64 + n in row-major M.
  for (int idx = threadIdx.x; idx < 4096; idx += 256) {
    const int k = idx >> 6, n = idx & 63;
    MshT[n * ES + k] = M[b * 4096 + idx];
  }

  // Build E tile: E[r, c] = c<32 ? cos(2pi*c*v) : sin(2pi*(c-32)*v)
  const float* xb = xnew + (size_t)b * 4096 * 2;
  float* Et = Esh[wave];
  for (int idx = lane; idx < 16 * 64; idx += 32) {
    const int   r = idx >> 6, c = idx & 63;
    const int   n = waveRow + r;
    const float v = xb[n * 2 + 1] * (31.0f / 32.0f);
    float t = (float)(c & 31) * v;
    t -= floorf(t);                               // exact range reduction to [0,1)
    const float ang = TWO_PI * t;
    Et[r * ES + c] = (c < 32) ? __cosf(ang) : __sinf(ang);
  }
  __syncthreads();

  // P(16x64) = E(16x64) * M(64x64) with f32 WMMA, K in chunks of 4.
  v8f c0 = {}, c1 = {}, c2 = {}, c3 = {};
  const int m16  = lane & 15;
  const int kOff = (lane >> 4) << 1;              // lanes 16-31 hold K=2,3
  const float* Ea  = &Et[m16 * ES + kOff];                 // A: lane = M row
  const float* Bt0 = &MshT[(m16 +  0) * ES + kOff];        // B: lane = N col
  const float* Bt1 = &MshT[(m16 + 16) * ES + kOff];
  const float* Bt2 = &MshT[(m16 + 32) * ES + kOff];
  const float* Bt3 = &MshT[(m16 + 48) * ES + kOff];
  #pragma unroll
  for (int k = 0; k < 64; k += 4) {
    const v2f a  = *(const v2f*)(Ea  + k);        // aligned ds_load_b64
    const v2f b0 = *(const v2f*)(Bt0 + k);
    const v2f b1 = *(const v2f*)(Bt1 + k);
    const v2f b2 = *(const v2f*)(Bt2 + k);
    const v2f b3 = *(const v2f*)(Bt3 + k);
    // 8 args: (neg_a, A, neg_b, B, c_mod, C, reuse_a, reuse_b)
    c0 = __builtin_amdgcn_wmma_f32_16x16x4_f32(false, a, false, b0, (short)0, c0, false, false);
    c1 = __builtin_amdgcn_wmma_f32_16x16x4_f32(false, a, false, b1, (short)0, c1, false, false);
    c2 = __builtin_amdgcn_wmma_f32_16x16x4_f32(false, a, false, b2, (short)0, c2, false, false);
    c3 = __builtin_amdgcn_wmma_f32_16x16x4_f32(false, a, false, b3, (short)0, c3, false, false);
  }

  // Spill P into this wave's private LDS tile (no barrier needed: wave-private,
  // same-wave LDS ordering handled via dscnt waits inserted by the compiler).
  {
    const int rlo = (lane < 16) ? 0 : 8;          // C/D layout: VGPR d -> M=d / d+8
    const int col = lane & 15;
    #pragma unroll
    for (int d = 0; d < 8; ++d) {
      const int r = rlo + d;
      Et[r * ES +  0 + col] = c0[d];
      Et[r * ES + 16 + col] = c1[d];
      Et[r * ES + 32 + col] = c2[d];
      Et[r * ES + 48 + col] = c3[d];
    }
  }

  // Epilogue: lanes (l, l^16) split the i-range of row r = l&15, then combine.
  {
    const int   r    = lane & 15;
    const int   half = lane >> 4;
    const int   n    = waveRow + r;
    const float u    = xb[n * 2 + 0] * (31.0f / 32.0f);
    float acc = 0.0f;
    #pragma unroll 4
    for (int i = half * 16; i < half * 16 + 16; ++i) {
      float t = (float)i * u;
      t -= floorf(t);
      const float ang = TWO_PI * t;
      acc += __cosf(ang) * Et[r * ES + i] - __sinf(ang) * Et[r * ES + 32 + i];
    }
    acc += __shfl_xor(acc, 16, 32);
    if (lane < 16)
      out[(size_t)b * 4096 + n] = acc * (1.0f / 1024.0f);
  }
}

// ---------------------------------------------------------------------------
// Launch: inputs per setup_inputs(): d_in[0]=y (8*32*32 f32), d_in[1]=xnew
// (8*64*64*2 f32). Output: 8*64*64 f32. Workspace: M = 8*64*64 f32 = 128 KB.
// ---------------------------------------------------------------------------
extern "C" void kernel_launch(void* const* d_in, const int* in_sizes, int n_in,
                              void* d_out, int out_size, void* d_ws, size_t ws_size,
                              hipStream_t stream) {
  (void)in_sizes; (void)n_in; (void)out_size; (void)ws_size;
  const float* y    = (const float*)d_in[0];
  const float* xnew = (const float*)d_in[1];
  float*       outp = (float*)d_out;
  float*       Mws  = (float*)d_ws;       // 8 * 4096 floats = 128 KB

  fi_fft_build_M<<<8, 1024, 0, stream>>>(y, Mws);
  fi_interp_wmma<<<dim3(32, 8), 256, 0, stream>>>(xnew, Mws, outp);
}